// Attention_58394375356576
// MI455X (gfx1250) — compile-verified
//
#include <hip/hip_runtime.h>

// Problem constants: B=32, S=128, IN=OUT=64, S*IN=8192.
// d_in order: h, W0, b0, Ws, bs, W1, b1 (all f32).

typedef float v2f __attribute__((ext_vector_type(2)));
typedef float v8f __attribute__((ext_vector_type(8)));

// D(16x16,f32) = A(16x4,f32) x B(4x16,f32) + C  -- full-precision CDNA5 WMMA
#define WMMA_F32x4(a, b, c) \
  __builtin_amdgcn_wmma_f32_16x16x4_f32(false, (a), false, (b), (short)0, (c), false, false)

// Workspace layout (float offsets). Total = 1,114,112 floats = 4.25 MB.
#define WS_WD      0        //  Wd[i][j]          64x64
#define WS_WEFF1   4096     //  Weff1[o][j]       64x64
#define WS_T3B1    8192     //  b1[o]+b0.Wsum[o]  64
#define WS_BIAST   16384    //  biasT[s][o]       128x64
#define WS_PART    24576    //  partial[16][b][o] 16x32x64
#define WS_T2      57344    //  Cagg[b][o]        32x64
#define WS_WEFF12  65536    //  (Weff1+Weff2)[s][o][j]  128x64x64
#define WS_WEFF3   589824   //  Weff3[s][o][j]          128x64x64

// ---------------------------------------------------------------------------
// K0: scalar prep (1 block, 256 threads).
//   Wsum[o][i] = sum_t W1[o, t*64+i]           (kept in LDS)
//   Wd[i][j]   = Ws[i][j] - W0a[i][j] - W0b[i][j]
//   Weff1[o][j]= sum_i Wsum[o][i] * W0a[i][j]
//   t3b1[o]    = b1[o] + sum_i b0[i]*Wsum[o][i]
// ---------------------------------------------------------------------------
__global__ void k0_prep(const float* __restrict__ W1, const float* __restrict__ W0,
                        const float* __restrict__ Wsm, const float* __restrict__ b0,
                        const float* __restrict__ b1, float* __restrict__ ws) {
  __shared__ float sWsum[64 * 64];
  const int tid = threadIdx.x;
  for (int e = tid; e < 4096; e += 256) {
    const int o = e >> 6, i = e & 63;
    const float* p = W1 + o * 8192 + i;
    float acc = 0.f;
#pragma unroll 8
    for (int t = 0; t < 128; ++t) acc += p[t * 64];
    sWsum[e] = acc;
  }
  __syncthreads();
  float* Wd    = ws + WS_WD;
  float* Weff1 = ws + WS_WEFF1;
  float* t3b1  = ws + WS_T3B1;
  for (int e = tid; e < 4096; e += 256) {
    const int i = e >> 6, j = e & 63;
    Wd[e] = Wsm[e] - W0[i * 128 + j] - W0[i * 128 + 64 + j];
  }
  for (int e = tid; e < 4096; e += 256) {
    const int o = e >> 6, j = e & 63;
    float acc = 0.f;
#pragma unroll 8
    for (int i = 0; i < 64; ++i) acc += sWsum[o * 64 + i] * W0[i * 128 + j];
    Weff1[e] = acc;
  }
  if (tid < 64) {
    float acc = b1[tid];
    for (int i = 0; i < 64; ++i) acc += b0[i] * sWsum[tid * 64 + i];
    t3b1[tid] = acc;
  }
}

// ---------------------------------------------------------------------------
// K1 (templated, grid=128 per s, block=256 = 8 waves):
//   dst[s][o][j] = sum_i W1[o, s*64+i] * Bsrc[i*BSTR + j]  (+ Weff1[o][j] if ADDW)
//  BSTR is a compile-time constant so every B load uses an immediate offset
//  from a fixed per-lane base pointer (no per-step 64-bit address math).
//  16 tiles / 8 waves = 2 tiles per wave; K=64 -> 16 wmma per tile.
// ---------------------------------------------------------------------------
template <int BSTR, bool ADDW, bool DOBIAS>
__global__ void k1_weights_t(const float* __restrict__ W1, const float* __restrict__ Bsrc,
                             const float* __restrict__ b0, const float* __restrict__ bs,
                             float* __restrict__ ws, float* __restrict__ dstAll) {
  const int s = blockIdx.x;
  const float* Weff1 = ws + WS_WEFF1;
  const float* t3b1  = ws + WS_T3B1;
  float* biasT = ws + WS_BIAST;
  float* dst   = dstAll + s * 4096;

  const int tid   = threadIdx.x;
  const int wave  = tid >> 5;
  const int lane  = tid & 31;
  const int lmod  = lane & 15;
  const int lhalf = lane >> 4;
  const float* A = W1 + s * 64;  // A[o][i] at A[o*8192+i]

#pragma unroll
  for (int q = 0; q < 2; ++q) {
    const int t16   = wave * 2 + q;     // wave-uniform tile id
    const int mBase = (t16 >> 2) * 16;  // o
    const int nBase = (t16 & 3) * 16;   // j
    const int n     = nBase + lmod;
    const float* Arow = A + (mBase + lmod) * 8192 + lhalf * 2;
    const float* Bp   = Bsrc + (lhalf * 2) * BSTR + n;
    v8f acc = {};
#pragma unroll
    for (int k = 0; k < 64; k += 4) {
      v2f a = *(const v2f*)(Arow + k);
      v2f b;
      b.x = Bp[k * BSTR];          // immediate offsets: BSTR is constexpr
      b.y = Bp[k * BSTR + BSTR];
      acc = WMMA_F32x4(a, b, acc);
    }
#pragma unroll
    for (int r = 0; r < 8; ++r) {
      const int m = mBase + r + lhalf * 8;
      float v = acc[r];
      if (ADDW) v += Weff1[m * 64 + n];
      dst[m * 64 + n] = v;
    }
  }
  // biasT[s][o] = t3b1[o] + sum_i (bs-b0)[i] * W1[o, s*64+i]
  if (DOBIAS && tid < 64) {
    const int o = tid;
    float acc = t3b1[o];
    const float* p = W1 + o * 8192 + s * 64;
    for (int i = 0; i < 64; ++i) acc += (bs[i] - b0[i]) * p[i];
    biasT[s * 64 + o] = acc;
  }
}

// ---------------------------------------------------------------------------
// K2: split-K GEMM for Cagg partials (grid=16, block=256 = 8 waves).
//  partial[kb][b][o] = sum_{t in [kb*8, kb*8+8)} sum_j h[b][t*64+j] * Weff3[t][o][j]
//  One 16x16 tile per wave (2x4 tiles cover 32x64), 128 wmma per wave.
// ---------------------------------------------------------------------------
__global__ void k2_cagg_partial(const float* __restrict__ h, float* __restrict__ ws) {
  const float* Weff3 = ws + WS_WEFF3;
  float* partial = ws + WS_PART + blockIdx.x * 2048;
  const int tid   = threadIdx.x;
  const int wave  = tid >> 5;
  const int lane  = tid & 31;
  const int lmod  = lane & 15;
  const int lhalf = lane >> 4;
  const int mBase = (wave >> 2) * 16;  // b
  const int nBase = (wave & 3) * 16;   // o
  const int n     = nBase + lmod;
  const int t0    = blockIdx.x * 8;

  const float* Arow0 = h + (mBase + lmod) * 8192 + t0 * 64 + lhalf * 2;
  const float* Brow0 = Weff3 + t0 * 4096 + n * 64 + lhalf * 2;
  v8f acc = {};
  for (int t = 0; t < 8; ++t) {
    const float* Arow = Arow0 + t * 64;
    const float* Brow = Brow0 + t * 4096;
    __builtin_prefetch(Arow + 64, 0, 1);    // gfx1250 global_prefetch_b8: next A t-block
    __builtin_prefetch(Brow + 4096, 0, 1);  // next Weff3 t-block
#pragma unroll
    for (int k = 0; k < 64; k += 4) {
      v2f a = *(const v2f*)(Arow + k);
      v2f b = *(const v2f*)(Brow + k);
      acc = WMMA_F32x4(a, b, acc);
    }
  }
#pragma unroll
  for (int r = 0; r < 8; ++r) {
    const int m = mBase + r + lhalf * 8;
    partial[m * 64 + n] = acc[r];
  }
}

// K2b: deterministic reduction of the 16 split-K partials into T2[b][o].
__global__ void k2b_reduce(float* __restrict__ ws) {
  const float* partial = ws + WS_PART;
  float* T2 = ws + WS_T2;
  for (int e = threadIdx.x; e < 2048; e += 256) {
    float acc = 0.f;
#pragma unroll
    for (int kb = 0; kb < 16; ++kb) acc += partial[kb * 2048 + e];
    T2[e] = acc;
  }
}

// ---------------------------------------------------------------------------
// K3: output (grid=128 per s, block=256 = 8 waves).
//  out[b][s][o] = sum_j h[b][s][j]*Weff12[s][o][j] + T2[b][o] + biasT[s][o]
// ---------------------------------------------------------------------------
__global__ void k3_out(const float* __restrict__ h, const float* __restrict__ ws,
                       float* __restrict__ out) {
  const int s = blockIdx.x;
  const float* biasT = ws + WS_BIAST;
  const float* T2    = ws + WS_T2;
  const float* Wt    = ws + WS_WEFF12 + s * 4096;  // [o][j]
  const int tid   = threadIdx.x;
  const int wave  = tid >> 5;
  const int lane  = tid & 31;
  const int lmod  = lane & 15;
  const int lhalf = lane >> 4;
  const int mBase = (wave >> 2) * 16;  // b
  const int nBase = (wave & 3) * 16;   // o
  const int n     = nBase + lmod;

  const float* Arow = h + (mBase + lmod) * 8192 + s * 64 + lhalf * 2;
  const float* Brow = Wt + n * 64 + lhalf * 2;
  v8f acc = {};
#pragma unroll
  for (int k = 0; k < 64; k += 4) {
    v2f a = *(const v2f*)(Arow + k);
    v2f b = *(const v2f*)(Brow + k);
    acc = WMMA_F32x4(a, b, acc);
  }
#pragma unroll
  for (int r = 0; r < 8; ++r) {
    const int m = mBase + r + lhalf * 8;  // b
    out[m * 8192 + s * 64 + n] = acc[r] + T2[m * 64 + n] + biasT[s * 64 + n];
  }
}

extern "C" void kernel_launch(void* const* d_in, const int* in_sizes, int n_in,
                              void* d_out, int out_size, void* d_ws, size_t ws_size,
                              hipStream_t stream) {
  const float* h   = (const float*)d_in[0];
  const float* W0  = (const float*)d_in[1];
  const float* b0  = (const float*)d_in[2];
  const float* Wsm = (const float*)d_in[3];
  const float* bs  = (const float*)d_in[4];
  const float* W1  = (const float*)d_in[5];
  const float* b1  = (const float*)d_in[6];
  float* ws  = (float*)d_ws;
  float* out = (float*)d_out;

  hipLaunchKernelGGL(k0_prep, dim3(1), dim3(256), 0, stream, W1, W0, Wsm, b0, b1, ws);
  // g==0: Weff12 = Weff1 + W1[s-block] x Wd   (B stride 64, +Weff1 epilogue)
  hipLaunchKernelGGL((k1_weights_t<64, true, false>), dim3(128), dim3(256), 0, stream,
                     W1, ws + WS_WD, b0, bs, ws, ws + WS_WEFF12);
  // g==1: Weff3 = W1[s-block] x W0b           (B stride 128, bias epilogue)
  hipLaunchKernelGGL((k1_weights_t<128, false, true>), dim3(128), dim3(256), 0, stream,
                     W1, W0 + 64, b0, bs, ws, ws + WS_WEFF3);
  hipLaunchKernelGGL(k2_cagg_partial, dim3(16), dim3(256), 0, stream, h, ws);
  hipLaunchKernelGGL(k2b_reduce,      dim3(1),  dim3(256), 0, stream, ws);
  hipLaunchKernelGGL(k3_out,          dim3(128), dim3(256), 0, stream, h, ws, out);
}